// GatedGraphConvEncoder_54546084659939
// MI455X (gfx1250) — compile-verified
//
#include <hip/hip_runtime.h>
#include <hip/hip_bf16.h>

#define NNODES   100000
#define CH       40
#define CH3      120
#define NLAYERS  40
#define NEDGES   1600000
#define NC       (NNODES * CH)        // 4,000,000 floats = 16 MB

typedef __attribute__((ext_vector_type(2))) float v2f;
typedef __attribute__((ext_vector_type(8))) float v8f;

// Exact-fp32 CDNA5 matrix op: V_WMMA_F32_16X16X4_F32 (D = A[16x4] * B[4x16] + C)
__device__ __forceinline__ v8f wmma4(v2f a, v2f b, v8f c) {
  return __builtin_amdgcn_wmma_f32_16x16x4_f32(
      /*neg_a=*/false, a, /*neg_b=*/false, b,
      /*c_mod=*/(short)0, c, /*reuse_a=*/false, /*reuse_b=*/false);
}

// ---------------------------------------------------------------- utilities
__global__ __launch_bounds__(256) void copy4_k(const float4* __restrict__ in,
                                               float4* __restrict__ out, int n4) {
  int i = blockIdx.x * 256 + threadIdx.x;
  if (i < n4) out[i] = in[i];
}

__global__ __launch_bounds__(256) void zero4_k(float4* __restrict__ p, int n4) {
  int i = blockIdx.x * 256 + threadIdx.x;
  if (i < n4) p[i] = make_float4(0.f, 0.f, 0.f, 0.f);
}

__global__ __launch_bounds__(256) void relu_k(float* __restrict__ p, int n) {
  int i = blockIdx.x * 256 + threadIdx.x;
  if (i < n) p[i] = fmaxf(p[i], 0.0f);
}

// ---------------------------------------------------------------- m = h @ W
// 32 nodes per block, 6 waves (192 threads): wave = (node_tile 0..1, col_tile 0..2)
__global__ __launch_bounds__(192)
void node_gemm40(const float* __restrict__ h, const float* __restrict__ W,
                 float* __restrict__ m) {
  __shared__ float sH[32][40];   // A tiles (node rows)
  __shared__ float sW[40][48];   // B, cols padded 40->48 with zeros
  const int tid  = threadIdx.x;
  const int base = blockIdx.x * 32;

  for (int i = tid; i < 40 * 48; i += 192) {
    int k = i / 48, n = i - k * 48;
    sW[k][n] = (n < CH) ? W[k * CH + n] : 0.0f;
  }
  for (int i = tid; i < 32 * 40; i += 192) {
    sH[i / 40][i % 40] = h[(long)base * CH + i];   // 32 contiguous rows
  }
  __syncthreads();

  const int wid  = tid >> 5;
  const int lane = tid & 31;
  const int row  = lane & 15;    // M for A / N for B
  const int half = lane >> 4;    // K half select
  const int mt   = wid / 3;      // node tile 0..1
  const int ct   = wid % 3;      // col tile 0..2

  v8f acc = {};
#pragma unroll
  for (int kk = 0; kk < 10; ++kk) {           // K = 40 = 10 x 4
    const int k0 = 4 * kk + 2 * half;
    v2f a; a.x = sH[mt * 16 + row][k0];       a.y = sH[mt * 16 + row][k0 + 1];
    v2f b; b.x = sW[k0][ct * 16 + row];       b.y = sW[k0 + 1][ct * 16 + row];
    acc = wmma4(a, b, acc);
  }

  const int ncol = ct * 16 + row;
  if (ncol < CH) {
#pragma unroll
    for (int v = 0; v < 8; ++v) {
      int mrow = base + mt * 16 + v + 8 * half;
      m[(long)mrow * CH + ncol] = acc[v];
    }
  }
}

// ------------------------------------------------- agg[dst] += m[src] (edges)
__global__ __launch_bounds__(256)
void scatter_add40(const float* __restrict__ m, const int* __restrict__ src,
                   const int* __restrict__ dst, float* __restrict__ agg) {
  long t = (long)blockIdx.x * 256 + threadIdx.x;   // over NEDGES*CH
  if (t >= (long)NEDGES * CH) return;
  int e = (int)(t / CH);
  int c = (int)(t - (long)e * CH);
  int s = src[e], d = dst[e];
  atomicAdd(&agg[(long)d * CH + c], m[(long)s * CH + c]);   // L2-resident f32 atomic
}

// -------------------------------------------------------------- GRU update
// 16 nodes per block, 8 waves: wave w computes col-tile w (of 128 padded cols)
// of BOTH gi = agg @ Wih^T and gh = h @ Whh^T, then gate math, h updated in place.
__global__ __launch_bounds__(256)
void gru_update(const float* __restrict__ agg, float* __restrict__ h,
                const float* __restrict__ Wih, const float* __restrict__ Whh,
                const float* __restrict__ bih, const float* __restrict__ bhh) {
  __shared__ float sBi[40][128];   // Wih^T, cols padded 120->128 with zeros
  __shared__ float sBh[40][128];   // Whh^T
  __shared__ float sA[16][40];     // agg rows
  __shared__ float sH[16][40];     // h rows
  __shared__ float sGi[16][128];   // gi result
  __shared__ float sGh[16][128];   // gh result
  const int tid  = threadIdx.x;
  const int base = blockIdx.x * 16;

  for (int i = tid; i < 40 * 128; i += 256) {
    int k = i >> 7, n = i & 127;
    sBi[k][n] = (n < CH3) ? Wih[n * CH + k] : 0.0f;   // transpose on stage-in
    sBh[k][n] = (n < CH3) ? Whh[n * CH + k] : 0.0f;
  }
  for (int i = tid; i < 16 * 40; i += 256) {
    sA[i / 40][i % 40] = agg[(long)base * CH + i];
    sH[i / 40][i % 40] = h[(long)base * CH + i];
  }
  __syncthreads();

  const int ct   = tid >> 5;     // col tile 0..7
  const int lane = tid & 31;
  const int row  = lane & 15;
  const int half = lane >> 4;

  v8f ai = {};
  v8f ah = {};
#pragma unroll
  for (int kk = 0; kk < 10; ++kk) {
    const int k0 = 4 * kk + 2 * half;
    v2f aA; aA.x = sA[row][k0];               aA.y = sA[row][k0 + 1];
    v2f aH; aH.x = sH[row][k0];               aH.y = sH[row][k0 + 1];
    v2f bI; bI.x = sBi[k0][ct * 16 + row];    bI.y = sBi[k0 + 1][ct * 16 + row];
    v2f bH; bH.x = sBh[k0][ct * 16 + row];    bH.y = sBh[k0 + 1][ct * 16 + row];
    ai = wmma4(aA, bI, ai);
    ah = wmma4(aH, bH, ah);
  }
#pragma unroll
  for (int v = 0; v < 8; ++v) {
    int mrow = v + 8 * half;
    sGi[mrow][ct * 16 + row] = ai[v];
    sGh[mrow][ct * 16 + row] = ah[v];
  }
  __syncthreads();

  for (int i = tid; i < 16 * 40; i += 256) {
    int nd = i / 40, c = i - nd * 40;
    float ir  = sGi[nd][c]      + bih[c];
    float hr  = sGh[nd][c]      + bhh[c];
    float iz  = sGi[nd][40 + c] + bih[40 + c];
    float hz  = sGh[nd][40 + c] + bhh[40 + c];
    float in_ = sGi[nd][80 + c] + bih[80 + c];
    float hn  = sGh[nd][80 + c] + bhh[80 + c];
    float r = 1.0f / (1.0f + __expf(-(ir + hr)));
    float z = 1.0f / (1.0f + __expf(-(iz + hz)));
    float nn = tanhf(in_ + r * hn);
    h[(long)base * CH + i] = (1.0f - z) * nn + z * sH[nd][c];
  }
}

// ---------------------------------------------------------------- launcher
extern "C" void kernel_launch(void* const* d_in, const int* in_sizes, int n_in,
                              void* d_out, int out_size, void* d_ws, size_t ws_size,
                              hipStream_t stream) {
  (void)in_sizes; (void)n_in; (void)out_size; (void)ws_size;

  const float* x    = (const float*)d_in[0];
  const float* W1   = (const float*)d_in[1];
  const float* Wih1 = (const float*)d_in[2];
  const float* Whh1 = (const float*)d_in[3];
  const float* bih1 = (const float*)d_in[4];
  const float* bhh1 = (const float*)d_in[5];
  const float* W2   = (const float*)d_in[6];
  const float* Wih2 = (const float*)d_in[7];
  const float* Whh2 = (const float*)d_in[8];
  const float* bih2 = (const float*)d_in[9];
  const float* bhh2 = (const float*)d_in[10];
  const int*   edge = (const int*)d_in[11];
  const int*   src  = edge;
  const int*   dst  = edge + NEDGES;

  float* h   = (float*)d_ws;        // 16 MB
  float* m   = h + NC;              // 16 MB
  float* agg = m + NC;              // 16 MB

  const int n4        = NC / 4;                       // 1,000,000 float4
  const int g4        = (n4 + 255) / 256;
  const int gRelu     = (NC + 255) / 256;             // 15625
  const int gGemm     = NNODES / 32;                  // 3125 (exact)
  const int gGru      = NNODES / 16;                  // 6250 (exact)
  const int gScatter  = (int)(((long)NEDGES * CH + 255) / 256);  // 250000

  // h = x
  copy4_k<<<g4, 256, 0, stream>>>((const float4*)x, (float4*)h, n4);

  for (int conv = 0; conv < 2; ++conv) {
    const float* W   = conv ? W2   : W1;
    const float* Wih = conv ? Wih2 : Wih1;
    const float* Whh = conv ? Whh2 : Whh1;
    const float* bih = conv ? bih2 : bih1;
    const float* bhh = conv ? bhh2 : bhh1;

    for (int l = 0; l < NLAYERS; ++l) {
      node_gemm40<<<gGemm, 192, 0, stream>>>(h, W + (long)l * CH * CH, m);
      zero4_k<<<g4, 256, 0, stream>>>((float4*)agg, n4);
      scatter_add40<<<gScatter, 256, 0, stream>>>(m, src, dst, agg);
      gru_update<<<gGru, 256, 0, stream>>>(agg, h, Wih, Whh, bih, bhh);
    }
    if (conv == 0) relu_k<<<gRelu, 256, 0, stream>>>(h, NC);
  }

  // out = h flattened
  copy4_k<<<g4, 256, 0, stream>>>((const float4*)h, (float4*)d_out, n4);
}